// ExternalMemory_47261820125704
// MI455X (gfx1250) — compile-verified
//
#include <hip/hip_runtime.h>
#include <math.h>

#define N_ROWS 131072
#define W 256

typedef __attribute__((ext_vector_type(2))) float v2f;
typedef __attribute__((ext_vector_type(4))) float v4f;
typedef __attribute__((ext_vector_type(8))) float v8f;

__device__ __forceinline__ float clamp01f(float x) {
    return fminf(fmaxf(x, 0.0f), 1.0f);
}

// ---------------------------------------------------------------------------
// Kernel Z: compute ||clamp01(key)|| into scal[2], zero accumulators scal[0..1]
// ---------------------------------------------------------------------------
__global__ __launch_bounds__(256) void ntm_init_kernel(const float* __restrict__ key,
                                                       float* __restrict__ scal) {
    __shared__ float red[256];
    int t = threadIdx.x;
    float k = clamp01f(key[t]);
    red[t] = k * k;
    __syncthreads();
    for (int off = 128; off; off >>= 1) {
        if (t < off) red[t] += red[t + off];
        __syncthreads();
    }
    if (t == 0) {
        scal[0] = 0.0f;             // sum of exp(beta*K)
        scal[1] = 0.0f;             // sum of wl^gamma
        scal[2] = sqrtf(red[0]);    // ||key_c||
    }
}

// ---------------------------------------------------------------------------
// Kernel A: content addressing.
// One wave32 per 16-row tile. Uses V_WMMA_F32_16X16X4_F32 for mem@key_c
// (key in B column 0), VALU for per-row sum-of-squares from the same loads.
// B operand built branchlessly: unconditional float2 LDS read, zeroed by a
// per-lane multiplicative flag (avoids exec-mask save/restore in hot loop).
// Writes e_arr[i] = exp(beta * cos_sim(mem[i], key_c)); accumulates sum.
// ---------------------------------------------------------------------------
__global__ __launch_bounds__(256) void ntm_content_kernel(const float* __restrict__ mem,
                                                          const float* __restrict__ key,
                                                          const float* __restrict__ beta_p,
                                                          float* __restrict__ e_arr,
                                                          float* __restrict__ scal) {
    __shared__ float kc[W];
    __shared__ float dots[8][16];

    int t = threadIdx.x;
    kc[t] = clamp01f(key[t]);
    __syncthreads();

    const int wave = t >> 5;
    const int lane = t & 31;
    const int half = lane >> 4;   // 0: K={k,k+1}, 1: K={k+2,k+3}
    const int m    = lane & 15;   // row within tile (A) / column (B)
    const float fm = (m == 0) ? 1.0f : 0.0f;  // key lives only in B column 0

    const long tile = (long)blockIdx.x * 8 + wave;
    const long row0 = tile * 16;
    const float* rowp = mem + (row0 + m) * (long)W;
    const float2* kc2 = (const float2*)kc;    // kc[col], kc[col+1] pairs

    v8f acc = {};          // C/D accumulator: D col 0 = dot products
    float sumsq = 0.0f;

#pragma unroll 8
    for (int k = 0; k < W; k += 4) {
        const int col = k + 2 * half;
        // A-matrix 16x4 f32 layout: V0 = K{0,2}, V1 = K{1,3} across lane halves
        float2 f = *(const float2*)(rowp + col);
        v2f aval;
        aval.x = f.x;
        aval.y = f.y;
        sumsq = fmaf(f.x, f.x, sumsq);
        sumsq = fmaf(f.y, f.y, sumsq);
        // B-matrix 4x16 f32: key chunk in column 0 (lanes 0 and 16), rest zero.
        // Unconditional LDS read (ds_load_b64) + flag multiply => branchless.
        float2 kp = kc2[col >> 1];
        v2f bval;
        bval.x = kp.x * fm;
        bval.y = kp.y * fm;
        acc = __builtin_amdgcn_wmma_f32_16x16x4_f32(
            /*neg_a=*/false, aval, /*neg_b=*/false, bval,
            /*c_mod=*/(short)0, acc, /*reuse_a=*/false, /*reuse_b=*/false);
    }

    // Row m total sum-of-squares lives split across lanes m and m+16.
    float rss = sumsq + __shfl_xor(sumsq, 16, 32);

    // D layout: lane 0 VGPRs 0..7 = rows 0..7 (col 0); lane 16 = rows 8..15.
    if (lane == 0) {
#pragma unroll
        for (int v = 0; v < 8; ++v) dots[wave][v] = acc[v];
    }
    if (lane == 16) {
#pragma unroll
        for (int v = 0; v < 8; ++v) dots[wave][8 + v] = acc[v];
    }
    __syncthreads();

    float ev = 0.0f;
    if (lane < 16) {
        float dot = dots[wave][m];
        float cosim = dot / (sqrtf(rss) * scal[2]);
        ev = expf(beta_p[0] * cosim);
        e_arr[row0 + m] = ev;
    }
    // wave-reduce ev (lanes >= 16 contribute 0)
    float s = ev;
#pragma unroll
    for (int off = 16; off; off >>= 1) s += __shfl_xor(s, off, 32);
    if (lane == 0) atomicAdd(&scal[0], s);
}

// ---------------------------------------------------------------------------
// Kernel BC: gate + circular shift + sharpen-pow.
// wg(i) recomputed locally for i-1, i, i+1 (pure function of e_arr, w_prev).
// ---------------------------------------------------------------------------
__global__ __launch_bounds__(256) void ntm_shift_pow_kernel(const float* __restrict__ e_arr,
                                                            const float* __restrict__ w_prev,
                                                            const float* __restrict__ gate_p,
                                                            const float* __restrict__ shift_p,
                                                            const float* __restrict__ gamma_p,
                                                            float* __restrict__ p_arr,
                                                            float* __restrict__ scal) {
    __shared__ float red[256];
    const int i = blockIdx.x * 256 + threadIdx.x;
    const float invS = 1.0f / scal[0];
    const float gate = gate_p[0];
    const float s0 = clamp01f(shift_p[0]);
    const float s1 = clamp01f(shift_p[1]);
    const float s2 = clamp01f(shift_p[2]);

    auto wgf = [&](int idx) -> float {
        float wc = e_arr[idx] * invS;
        float wp = clamp01f(w_prev[idx]);
        return gate * wc + (1.0f - gate) * wp;
    };

    const int ip = (i + 1) & (N_ROWS - 1);
    const int im = (i + N_ROWS - 1) & (N_ROWS - 1);
    // wl = s0 * roll(wg,-1) + s1 * wg + s2 * roll(wg,+1)
    float wl = s0 * wgf(ip) + s1 * wgf(i) + s2 * wgf(im);
    float p = powf(wl, gamma_p[0]);
    p_arr[i] = p;

    // block-reduce p -> one atomicAdd per block
    int t = threadIdx.x;
    red[t] = p;
    __syncthreads();
    for (int off = 128; off; off >>= 1) {
        if (t < off) red[t] += red[t + off];
        __syncthreads();
    }
    if (t == 0) atomicAdd(&scal[1], red[0]);
}

// ---------------------------------------------------------------------------
// Kernel D: sharpen normalize -> final address weights w (output 3)
// ---------------------------------------------------------------------------
__global__ __launch_bounds__(256) void ntm_norm_kernel(const float* __restrict__ p_arr,
                                                       const float* __restrict__ scal,
                                                       float* __restrict__ w_out) {
    const int i = blockIdx.x * 256 + threadIdx.x;
    w_out[i] = p_arr[i] / scal[1];
}

// ---------------------------------------------------------------------------
// Kernel E: fused read + erase + add. Streams mem (16B vectors, coalesced;
// the 128 MiB matrix fits in the 192 MB L2, so this read should hit lines
// left by kernel A). new_mem is write-once/never-re-read -> non-temporal
// stores (native clang vector type for the builtin) so 128 MiB of dead store
// data does not evict mem from L2. Accumulates per-block partials of
// r = mem^T @ w_r. 64 rows per block; thread t: col4 = t&63, subgroup = t>>6.
// ---------------------------------------------------------------------------
#define ROWS_PER_BLOCK 64

__global__ __launch_bounds__(256) void ntm_update_kernel(const float* __restrict__ mem,
                                                         const float* __restrict__ e_vec,
                                                         const float* __restrict__ a_vec,
                                                         const float* __restrict__ w_out,
                                                         float* __restrict__ new_mem,
                                                         float* __restrict__ r_partials) {
    __shared__ float wr_s[ROWS_PER_BLOCK];
    __shared__ float red[256 * 4];

    const int t = threadIdx.x;
    const int j4 = t & 63;         // which 16B vector of the 256-wide row
    const int rsub = t >> 6;       // 0..3
    const long rowBase = (long)blockIdx.x * ROWS_PER_BLOCK;

    v4f e4 = ((const v4f*)e_vec)[j4];
    v4f a4 = ((const v4f*)a_vec)[j4];
    e4.x = clamp01f(e4.x); e4.y = clamp01f(e4.y); e4.z = clamp01f(e4.z); e4.w = clamp01f(e4.w);
    a4.x = clamp01f(a4.x); a4.y = clamp01f(a4.y); a4.z = clamp01f(a4.z); a4.w = clamp01f(a4.w);

    if (t < ROWS_PER_BLOCK) wr_s[t] = clamp01f(w_out[rowBase + t]);
    __syncthreads();

    float rx = 0.0f, ry = 0.0f, rz = 0.0f, rw = 0.0f;
#pragma unroll 4
    for (int rr = rsub; rr < ROWS_PER_BLOCK; rr += 4) {
        const long row = rowBase + rr;
        const float wr = wr_s[rr];
        const v4f m4 = ((const v4f*)(mem + row * (long)W))[j4];
        v4f nm;
        nm.x = m4.x * (1.0f - wr * e4.x) + wr * a4.x;
        nm.y = m4.y * (1.0f - wr * e4.y) + wr * a4.y;
        nm.z = m4.z * (1.0f - wr * e4.z) + wr * a4.z;
        nm.w = m4.w * (1.0f - wr * e4.w) + wr * a4.w;
        __builtin_nontemporal_store(nm, (v4f*)(new_mem + row * (long)W) + j4);
        rx = fmaf(m4.x, wr, rx);
        ry = fmaf(m4.y, wr, ry);
        rz = fmaf(m4.z, wr, rz);
        rw = fmaf(m4.w, wr, rw);
    }

    red[t * 4 + 0] = rx;
    red[t * 4 + 1] = ry;
    red[t * 4 + 2] = rz;
    red[t * 4 + 3] = rw;
    __syncthreads();
    if (t < 64) {
        v4f s;
        s.x = red[t*4+0] + red[(t+64)*4+0] + red[(t+128)*4+0] + red[(t+192)*4+0];
        s.y = red[t*4+1] + red[(t+64)*4+1] + red[(t+128)*4+1] + red[(t+192)*4+1];
        s.z = red[t*4+2] + red[(t+64)*4+2] + red[(t+128)*4+2] + red[(t+192)*4+2];
        s.w = red[t*4+3] + red[(t+64)*4+3] + red[(t+128)*4+3] + red[(t+192)*4+3];
        ((v4f*)(r_partials + (long)blockIdx.x * W))[t] = s;
    }
}

// ---------------------------------------------------------------------------
// Kernel F: deterministic reduction of r partials -> r output
// ---------------------------------------------------------------------------
__global__ __launch_bounds__(256) void ntm_rred_kernel(const float* __restrict__ r_partials,
                                                       float* __restrict__ r_out,
                                                       int nblocks) {
    const int j = threadIdx.x;
    float s = 0.0f;
    for (int b = 0; b < nblocks; ++b) s += r_partials[(long)b * W + j];
    r_out[j] = s;
}

// ---------------------------------------------------------------------------
extern "C" void kernel_launch(void* const* d_in, const int* in_sizes, int n_in,
                              void* d_out, int out_size, void* d_ws, size_t ws_size,
                              hipStream_t stream) {
    const float* mem   = (const float*)d_in[0];
    const float* key   = (const float*)d_in[1];
    const float* beta  = (const float*)d_in[2];
    const float* gamma = (const float*)d_in[3];
    const float* gate  = (const float*)d_in[4];
    const float* shift = (const float*)d_in[5];
    const float* wprev = (const float*)d_in[6];
    const float* evec  = (const float*)d_in[7];
    const float* avec  = (const float*)d_in[8];

    float* r_out  = (float*)d_out;                       // [W]
    float* nm_out = r_out + W;                           // [N_ROWS*W]
    float* w_out  = nm_out + (size_t)N_ROWS * W;         // [N_ROWS]

    float* scal   = (float*)d_ws;                        // 16 floats reserved
    float* e_arr  = scal + 16;                           // [N_ROWS]
    float* p_arr  = e_arr + N_ROWS;                      // [N_ROWS]
    float* r_part = p_arr + N_ROWS;                      // [(N/64)*W]

    ntm_init_kernel<<<1, 256, 0, stream>>>(key, scal);
    ntm_content_kernel<<<N_ROWS / 128, 256, 0, stream>>>(mem, key, beta, e_arr, scal);
    ntm_shift_pow_kernel<<<N_ROWS / 256, 256, 0, stream>>>(e_arr, wprev, gate, shift, gamma,
                                                           p_arr, scal);
    ntm_norm_kernel<<<N_ROWS / 256, 256, 0, stream>>>(p_arr, scal, w_out);
    ntm_update_kernel<<<N_ROWS / ROWS_PER_BLOCK, 256, 0, stream>>>(mem, evec, avec, w_out,
                                                                   nm_out, r_part);
    ntm_rred_kernel<<<1, 256, 0, stream>>>(r_part, r_out, N_ROWS / ROWS_PER_BLOCK);
}